// QFCModel2_2697239462487
// MI455X (gfx1250) — compile-verified
//
#include <hip/hip_runtime.h>
#include <hip/hip_bf16.h>

// ---------------------------------------------------------------------------
// 4-qubit circuit, B=524288 batch.
// encoded state = tensor product of 4 single-qubit states (per-element angles)
// fixed tail gates fold into one 16x16 complex matrix M (prep kernel).
// final = M * v  -> done as f32 WMMA (V_WMMA_F32_16X16X4_F32), K=32 chain:
//   Re = [Mr | -Mi] [Vr;Vi],  Im = [Mi | Mr] [Vr;Vi]
// ---------------------------------------------------------------------------

struct Cplx { float r, i; };
__device__ __host__ inline Cplx cmul(Cplx a, Cplx b) {
    return { a.r * b.r - a.i * b.i, a.r * b.i + a.i * b.r };
}
__device__ __host__ inline Cplx cadd(Cplx a, Cplx b) { return { a.r + b.r, a.i + b.i }; }

typedef __attribute__((ext_vector_type(2))) float v2f;
typedef __attribute__((ext_vector_type(8))) float v8f;

// ---------------------------------------------------------------------------
// Prep: compose fixed gates into M (16x16 complex), store Mr, Mi to ws.
// Wire w <-> state bit (3-w) (wire 0 is most significant, C-order reshape).
// ---------------------------------------------------------------------------
__global__ void qfc_prep_kernel(const float* t_rx, const float* t_ry,
                                const float* t_rz, const float* t_crx,
                                float* __restrict__ ws) {
    if (threadIdx.x != 0 || blockIdx.x != 0) return;

    Cplx M[16][16];
    for (int i = 0; i < 16; ++i)
        for (int j = 0; j < 16; ++j)
            M[i][j] = { (i == j) ? 1.0f : 0.0f, 0.0f };

    auto ap1 = [&](int w, Cplx g00, Cplx g01, Cplx g10, Cplx g11) {
        int stride = 1 << (3 - w);
        for (int s0 = 0; s0 < 16; ++s0) {
            if (s0 & stride) continue;
            int s1 = s0 | stride;
            for (int c = 0; c < 16; ++c) {
                Cplx a = M[s0][c], b = M[s1][c];
                M[s0][c] = cadd(cmul(g00, a), cmul(g01, b));
                M[s1][c] = cadd(cmul(g10, a), cmul(g11, b));
            }
        }
    };

    float h, c, s;
    // rx(t_rx) on wire 0
    h = 0.5f * t_rx[0]; c = __cosf(h); s = __sinf(h);
    ap1(0, {c,0}, {0,-s}, {0,-s}, {c,0});
    // ry(t_ry) on wire 1
    h = 0.5f * t_ry[0]; c = __cosf(h); s = __sinf(h);
    ap1(1, {c,0}, {-s,0}, {s,0}, {c,0});
    // rz(t_rz) on wire 3
    h = 0.5f * t_rz[0]; c = __cosf(h); s = __sinf(h);
    ap1(3, {c,-s}, {0,0}, {0,0}, {c,s});
    // CRX on (wire0 ctrl = bit3, wire2 tgt = bit1)
    h = 0.5f * t_crx[0]; c = __cosf(h); s = __sinf(h);
    for (int s0 = 8; s0 < 16; ++s0) {
        if (s0 & 2) continue;
        int s1 = s0 | 2;
        for (int col = 0; col < 16; ++col) {
            Cplx a = M[s0][col], b = M[s1][col];
            M[s0][col] = { c * a.r + s * b.i,  c * a.i - s * b.r };   // c*a + (-i s)*b
            M[s1][col] = { s * a.i + c * b.r, -s * a.r + c * b.i };   // (-i s)*a + c*b
        }
    }
    // H on wire 3
    const float r2 = 0.70710678118654752f;
    ap1(3, {r2,0}, {r2,0}, {r2,0}, {-r2,0});
    // SX on wire 2
    ap1(2, {0.5f,0.5f}, {0.5f,-0.5f}, {0.5f,-0.5f}, {0.5f,0.5f});
    // CNOT (wire3 ctrl = bit0, wire0 tgt = bit3): if q3==1 swap q0 pair
    for (int s0 = 1; s0 < 8; s0 += 2) {   // q3=1, q0=0 : {1,3,5,7}
        int s1 = s0 | 8;
        for (int col = 0; col < 16; ++col) {
            Cplx t = M[s0][col]; M[s0][col] = M[s1][col]; M[s1][col] = t;
        }
    }
    // U on (wire1 = bit2, wire2 = bit1): new|10> = i*|11>, new|11> = -i*|10>
    for (int sidx = 0; sidx < 16; ++sidx) {
        if (!(sidx & 4) || (sidx & 2)) continue;   // q1=1, q2=0 : {4,5,12,13}
        int s10 = sidx, s11 = sidx | 2;
        for (int col = 0; col < 16; ++col) {
            Cplx a = M[s10][col], b = M[s11][col];
            M[s10][col] = { -b.i,  b.r };   //  i * b
            M[s11][col] = {  a.i, -a.r };   // -i * a
        }
    }

    float* Mr = ws;
    float* Mi = ws + 256;
    for (int i = 0; i < 16; ++i)
        for (int j = 0; j < 16; ++j) {
            Mr[i * 16 + j] = M[i][j].r;
            Mi[i * 16 + j] = M[i][j].i;
        }
}

// ---------------------------------------------------------------------------
// Main: one wave32 per 16-element tile. Lane n=l%16 <-> batch column,
// lane half h=l/16 <-> K sub-stripe (operand layout per ISA 7.12.2 tables:
// 32-bit frag VGPR r at lane-half h holds K = 4q + r + 2h).
// ---------------------------------------------------------------------------
__global__ __launch_bounds__(256) void qfc_main_kernel(
        const float* __restrict__ x, const float* __restrict__ inj,
        const float* __restrict__ ws, float* __restrict__ out) {
    const int lane = threadIdx.x & 31;
    const int wave = threadIdx.x >> 5;
    const int n    = lane & 15;          // batch column in tile
    const int hh   = lane >> 4;          // lane half
    const bool hi  = (hh != 0);
    const long tile = (long)blockIdx.x * 8 + wave;
    const long elem = tile * 16 + n;

    // ---- load angles (128B/lane, b128 loads) -----------------------------
    const float4* x4 = reinterpret_cast<const float4*>(x + elem * 16);
    const float4* i4 = reinterpret_cast<const float4*>(inj + elem * 16);
    float ang[16];
    #pragma unroll
    for (int q = 0; q < 4; ++q) {
        float4 a = x4[q], b = i4[q];
        ang[4*q+0] = a.x + b.x; ang[4*q+1] = a.y + b.y;
        ang[4*q+2] = a.z + b.z; ang[4*q+3] = a.w + b.w;
    }

    // ---- single-qubit states: lane computes wires {2h, 2h+1}, partner rest
    // psi_w = Rx(ang[12+w]) Rz(ang[8+w]) Ry(ang[4+w]) Rx(ang[w]) |0>
    Cplx mine[2][2];   // mine[t][comp], wire = 2h+t
    #pragma unroll
    for (int t = 0; t < 2; ++t) {
        int w = 2 * hh + t;
        float sa, ca, sb, cb, sz, cz, sd, cd;
        __sincosf(0.5f * ang[w],      &sa, &ca);
        __sincosf(0.5f * ang[4 + w],  &sb, &cb);
        __sincosf(0.5f * ang[8 + w],  &sz, &cz);
        __sincosf(0.5f * ang[12 + w], &sd, &cd);
        Cplx a0 = { cb * ca,  sb * sa };         // after Rx, Ry
        Cplx a1 = { sb * ca, -cb * sa };
        a0 = cmul(a0, { cz, -sz });              // Rz
        a1 = cmul(a1, { cz,  sz });
        mine[t][0] = { cd * a0.r + sd * a1.i,  cd * a0.i - sd * a1.r };  // Rx(d)
        mine[t][1] = { sd * a0.i + cd * a1.r, -sd * a0.r + cd * a1.i };
    }

    // exchange with partner lane (l ^ 16)
    Cplx recv[2][2];
    #pragma unroll
    for (int t = 0; t < 2; ++t)
        #pragma unroll
        for (int cidx = 0; cidx < 2; ++cidx) {
            recv[t][cidx].r = __shfl_xor(mine[t][cidx].r, 16, 32);
            recv[t][cidx].i = __shfl_xor(mine[t][cidx].i, 16, 32);
        }

    // assemble: psi0,psi1 (full), psi2[h] (single component), psi3 (full)
    Cplx P0_0 = hi ? recv[0][0] : mine[0][0];
    Cplx P0_1 = hi ? recv[0][1] : mine[0][1];
    Cplx P1_0 = hi ? recv[1][0] : mine[1][0];
    Cplx P1_1 = hi ? recv[1][1] : mine[1][1];
    Cplx P2h  = hi ? mine[0][1] : recv[0][0];
    Cplx P3_0 = hi ? mine[1][0] : recv[1][0];
    Cplx P3_1 = hi ? mine[1][1] : recv[1][1];

    // tensor product amplitudes for s = 4q + 2h + q3  (q = 2*q0 + q1)
    Cplx t23_0 = cmul(P2h, P3_0), t23_1 = cmul(P2h, P3_1);
    Cplx t01[4] = { cmul(P0_0, P1_0), cmul(P0_0, P1_1),
                    cmul(P0_1, P1_0), cmul(P0_1, P1_1) };
    Cplx amp[8];
    #pragma unroll
    for (int q = 0; q < 4; ++q) {
        amp[2*q]     = cmul(t01[q], t23_0);
        amp[2*q + 1] = cmul(t01[q], t23_1);
    }

    // B fragments: chunk q<4 -> Re, chunk q+4 -> Im  (K = s, 16+s)
    v2f bfr[8];
    #pragma unroll
    for (int q = 0; q < 4; ++q) {
        bfr[q]     = v2f{ amp[2*q].r, amp[2*q + 1].r };
        bfr[q + 4] = v2f{ amp[2*q].i, amp[2*q + 1].i };
    }

    // A fragments from ws (row m = n, cols k0 = 4q+2h, k0+1), float2 loads
    const float* Mr = ws;
    const float* Mi = ws + 256;
    v2f mrq[4], miq[4], nmi[4];
    #pragma unroll
    for (int q = 0; q < 4; ++q) {
        int off = n * 16 + 4 * q + 2 * hh;
        mrq[q] = *reinterpret_cast<const v2f*>(Mr + off);
        miq[q] = *reinterpret_cast<const v2f*>(Mi + off);
        nmi[q] = -miq[q];
    }

    // ---- WMMA chains: Re = Mr*Vr - Mi*Vi ; Im = Mi*Vr + Mr*Vi ------------
    v8f accr = { 0.f, 0.f, 0.f, 0.f, 0.f, 0.f, 0.f, 0.f };
    v8f acci = { 0.f, 0.f, 0.f, 0.f, 0.f, 0.f, 0.f, 0.f };
    #pragma unroll
    for (int q = 0; q < 4; ++q) {
        accr = __builtin_amdgcn_wmma_f32_16x16x4_f32(false, mrq[q], false, bfr[q],
                                                     (short)0, accr, false, false);
        acci = __builtin_amdgcn_wmma_f32_16x16x4_f32(false, miq[q], false, bfr[q],
                                                     (short)0, acci, false, false);
    }
    #pragma unroll
    for (int q = 0; q < 4; ++q) {
        accr = __builtin_amdgcn_wmma_f32_16x16x4_f32(false, nmi[q], false, bfr[q + 4],
                                                     (short)0, accr, false, false);
        acci = __builtin_amdgcn_wmma_f32_16x16x4_f32(false, mrq[q], false, bfr[q + 4],
                                                     (short)0, acci, false, false);
    }

    // ---- measurement: lane holds rows s = r + 8h of its column ------------
    // y0 = sum (z0+z1)|f|^2 ; y1 = sum (z2+z3)|f|^2 ; q0 = h
    float g0h = hi ? -1.0f : 1.0f;
    float y0p = 0.f, y1p = 0.f;
    #pragma unroll
    for (int r = 0; r < 8; ++r) {
        float p = accr[r] * accr[r] + acci[r] * acci[r];
        const int q1 = (r >> 2) & 1, q2 = (r >> 1) & 1, q3 = r & 1;
        y0p += (g0h + (float)(1 - 2 * q1)) * p;
        y1p += ((float)(1 - 2 * q2) + (float)(1 - 2 * q3)) * p;
    }
    float y0 = y0p + __shfl_xor(y0p, 16, 32);
    float y1 = y1p + __shfl_xor(y1p, 16, 32);

    // ---- upsample x8: out[elem][0..7] = y0, out[elem][8..15] = y1 --------
    float val = hi ? y1 : y0;
    float4 v4 = { val, val, val, val };
    float4* o = reinterpret_cast<float4*>(out + elem * 16 + 8 * hh);
    o[0] = v4;
    o[1] = v4;
}

// ---------------------------------------------------------------------------
extern "C" void kernel_launch(void* const* d_in, const int* in_sizes, int n_in,
                              void* d_out, int out_size, void* d_ws, size_t ws_size,
                              hipStream_t stream) {
    const float* x     = (const float*)d_in[0];
    const float* inj   = (const float*)d_in[1];
    const float* t_rx  = (const float*)d_in[2];
    const float* t_ry  = (const float*)d_in[3];
    const float* t_rz  = (const float*)d_in[4];
    const float* t_crx = (const float*)d_in[5];
    float* out = (float*)d_out;
    float* ws  = (float*)d_ws;

    const int B = in_sizes[0] / 16;          // 524288
    // 1) fold fixed gates into 16x16 complex matrix in workspace (2 KB)
    qfc_prep_kernel<<<1, 32, 0, stream>>>(t_rx, t_ry, t_rz, t_crx, ws);
    // 2) 8 waves/block, 16 batch elements per wave -> 128 per block
    const int blocks = B / 128;              // B divisible (4096 blocks)
    qfc_main_kernel<<<blocks, 256, 0, stream>>>(x, inj, ws, out);
}